// AveragePrevEmbeddingsLM_55044300865741
// MI455X (gfx1250) — compile-verified
//
#include <hip/hip_runtime.h>
#include <hip/hip_bf16.h>
#include <stddef.h>

// Problem constants (match reference)
#define VOCAB   32000
#define EMB     64
#define BATCH   4
#define SEQ     2048
#define MTOT    (BATCH * SEQ)   // 8192
#define NCH     16              // scan chunks per batch row
#define CHUNK   (SEQ / NCH)     // 128

typedef __attribute__((ext_vector_type(2))) float v2f;
typedef __attribute__((ext_vector_type(8))) float v8f;

// ---------------------------------------------------------------------------
// Phase 1: per-chunk channel sums  S[b][ch][e] = sum_{t in chunk} emb[x[b,t]][e]
// ---------------------------------------------------------------------------
__global__ __launch_bounds__(EMB)
void chunk_sum_kernel(const float* __restrict__ emb, const int* __restrict__ x,
                      float* __restrict__ S) {
    const int e  = threadIdx.x;          // 0..63 channel
    const int bc = blockIdx.x;           // b*NCH + ch
    const int b  = bc / NCH;
    const int ch = bc % NCH;
    const int* xs = x + b * SEQ + ch * CHUNK;
    float s = 0.0f;
    #pragma unroll 4
    for (int t = 0; t < CHUNK; ++t) {
        s += emb[(size_t)xs[t] * EMB + e];
    }
    S[(size_t)bc * EMB + e] = s;
}

// ---------------------------------------------------------------------------
// Phase 2: chunked inclusive scan + divide -> prefix means Y (row-major MTOT x 64)
// ---------------------------------------------------------------------------
__global__ __launch_bounds__(EMB)
void prefix_mean_kernel(const float* __restrict__ emb, const int* __restrict__ x,
                        const float* __restrict__ S, float* __restrict__ Y) {
    const int e  = threadIdx.x;
    const int bc = blockIdx.x;
    const int b  = bc / NCH;
    const int ch = bc % NCH;

    float carry = 0.0f;
    for (int c = 0; c < ch; ++c)
        carry += S[(size_t)(b * NCH + c) * EMB + e];

    const int t0 = ch * CHUNK;
    const int* xs = x + b * SEQ + t0;
    float* Yp = Y + ((size_t)b * SEQ + t0) * EMB + e;
    for (int t = 0; t < CHUNK; ++t) {
        carry += emb[(size_t)xs[t] * EMB + e];
        Yp[(size_t)t * EMB] = carry * (1.0f / (float)(t0 + t + 1));
    }
}

// ---------------------------------------------------------------------------
// Phase 3: logits = Y(8192x64) @ W(32000x64)^T + bias, via V_WMMA_F32_16X16X4_F32
// Block = 128 threads (4 waves). Each wave: one 32x32 C tile.
// grid.x indexes N in steps of 128 (4 waves * 32), grid.y indexes M in steps of 32.
// Output stores are non-temporal: the 1.05 GB logits stream must not evict the
// L2-resident W (8.2 MB) / Y (2 MB) operands that every M-tile pass re-reads.
// ---------------------------------------------------------------------------
__global__ __launch_bounds__(128)
void gemm_wmma_kernel(const float* __restrict__ Y, const float* __restrict__ W,
                      const float* __restrict__ bias, float* __restrict__ out) {
    const int wave = threadIdx.x >> 5;
    const int lane = threadIdx.x & 31;
    const int half = lane >> 4;      // 0: lanes 0-15, 1: lanes 16-31
    const int l    = lane & 15;

    const int n0 = (blockIdx.x * 4 + wave) * 32;   // N tile base (two 16-wide subtiles)
    const int m0 = blockIdx.y * 32;                // M tile base (two 16-tall subtiles)

    // A operand rows (from Y), B operand rows (from W). Per WMMA k-step s:
    //   lanes 0-15 hold K = 4s+0 / 4s+1, lanes 16-31 hold K = 4s+2 / 4s+3
    // => contiguous float2 at offset 4s + 2*half in the 64-long K dimension.
    const float* Ya0 = Y + (size_t)(m0 + l)      * EMB + 2 * half;
    const float* Ya1 = Y + (size_t)(m0 + 16 + l) * EMB + 2 * half;
    const float* Wb0 = W + (size_t)(n0 + l)      * EMB + 2 * half;
    const float* Wb1 = W + (size_t)(n0 + 16 + l) * EMB + 2 * half;

    // Accumulator init = bias (depends only on N / lane).
    const float b0 = bias[n0 + l];
    const float b1 = bias[n0 + 16 + l];
    v8f acc00, acc01, acc10, acc11;
    #pragma unroll
    for (int i = 0; i < 8; ++i) {
        acc00[i] = b0; acc01[i] = b1; acc10[i] = b0; acc11[i] = b1;
    }

    #pragma unroll
    for (int s = 0; s < 16; ++s) {               // K = 64 = 16 steps of 4
        const v2f a0 = *(const v2f*)(Ya0 + 4 * s);
        const v2f a1 = *(const v2f*)(Ya1 + 4 * s);
        const v2f w0 = *(const v2f*)(Wb0 + 4 * s);
        const v2f w1 = *(const v2f*)(Wb1 + 4 * s);
        acc00 = __builtin_amdgcn_wmma_f32_16x16x4_f32(false, a0, false, w0,
                                                      (short)0, acc00, false, false);
        acc01 = __builtin_amdgcn_wmma_f32_16x16x4_f32(false, a0, false, w1,
                                                      (short)0, acc01, false, false);
        acc10 = __builtin_amdgcn_wmma_f32_16x16x4_f32(false, a1, false, w0,
                                                      (short)0, acc10, false, false);
        acc11 = __builtin_amdgcn_wmma_f32_16x16x4_f32(false, a1, false, w1,
                                                      (short)0, acc11, false, false);
    }

    // C/D layout: VGPR v -> row m0+v (lanes 0-15) or m0+v+8 (lanes 16-31), col n0+l.
    // Non-temporal stores: streamed output, never re-read by this dispatch.
    #pragma unroll
    for (int v = 0; v < 8; ++v) {
        const size_t mA = (size_t)(m0 + v + half * 8);
        const size_t mB = mA + 16;
        __builtin_nontemporal_store(acc00[v], out + mA * VOCAB + n0 + l);
        __builtin_nontemporal_store(acc01[v], out + mA * VOCAB + n0 + 16 + l);
        __builtin_nontemporal_store(acc10[v], out + mB * VOCAB + n0 + l);
        __builtin_nontemporal_store(acc11[v], out + mB * VOCAB + n0 + 16 + l);
    }
}

// ---------------------------------------------------------------------------
extern "C" void kernel_launch(void* const* d_in, const int* in_sizes, int n_in,
                              void* d_out, int out_size, void* d_ws, size_t ws_size,
                              hipStream_t stream) {
    const float* emb  = (const float*)d_in[0];   // (32000, 64)
    const float* W    = (const float*)d_in[1];   // (32000, 64)
    const float* bias = (const float*)d_in[2];   // (32000,)
    const int*   x    = (const int*)d_in[3];     // (4, 2048)
    float* out = (float*)d_out;                  // (4, 2048, 32000)

    // Workspace layout: S (chunk sums) then Y (prefix means)
    float* S = (float*)d_ws;                                             // 16 KB
    float* Y = (float*)((char*)d_ws + (size_t)BATCH * NCH * EMB * sizeof(float)); // 2 MB

    chunk_sum_kernel<<<BATCH * NCH, EMB, 0, stream>>>(emb, x, S);
    prefix_mean_kernel<<<BATCH * NCH, EMB, 0, stream>>>(emb, x, S, Y);

    dim3 grid(VOCAB / 128, MTOT / 32);   // (250, 256)
    gemm_wmma_kernel<<<grid, 128, 0, stream>>>(Y, W, bias, out);
}